// HandshakingKerner_15436112462121
// MI455X (gfx1250) — compile-verified
//
#include <hip/hip_runtime.h>
#include <cmath>

typedef __attribute__((ext_vector_type(2))) float v2f;
typedef __attribute__((ext_vector_type(4))) float v4f;
typedef __attribute__((ext_vector_type(8))) float v8f;

#define Bn 8
#define Sn 192
#define Dn 768
#define Hn 384
#define Tn (Sn * (Sn + 1) / 2)   // 18528 pairs
#define Mn (Bn * Sn)             // 1536 rows of X

// ---------------------------------------------------------------------------
// Kernel 1: P = X @ W[:D], Q = X @ W[D:]  via V_WMMA_F32_16X16X4_F32.
// One wave computes a 32(M) x 16(N) tile of BOTH P and Q: the B fragment
// (2 strided b32 loads, the expensive operand) is reused across two M-tiles
// and both output matrices share the A fragment -> 1.5 load-instr per WMMA.
// ---------------------------------------------------------------------------
__global__ __launch_bounds__(32)
void pairproj_gemm_wmma(const float* __restrict__ X,   // [Mn, Dn]
                        const float* __restrict__ W,   // [2*Dn, Hn]
                        float* __restrict__ P,         // [Mn, Hn]
                        float* __restrict__ Q) {       // [Mn, Hn]
  const int m0   = blockIdx.x * 32;    // two 16-row M-tiles
  const int n0   = blockIdx.y * 16;
  const int lane = threadIdx.x;        // 0..31, wave32
  const int half = lane >> 4;          // 0: K lo pair, 1: K hi pair
  const int l15  = lane & 15;

  const int bcol = n0 + l15;           // B/C/D: N striped over lanes

  // A fragment: .x = X[row, k + 2*half], .y = X[row, k + 2*half + 1]
  const float* aPtr0 = X + (size_t)(m0 + l15) * Dn + 2 * half;
  const float* aPtr1 = aPtr0 + (size_t)16 * Dn;
  // B fragment: element [K = k + 2*half + {0,1}, N = bcol]
  const float* bPtrP = W + (size_t)(2 * half) * Hn + bcol;
  const float* bPtrQ = bPtrP + (size_t)Dn * Hn;

  v8f cp0 = {}, cq0 = {}, cp1 = {}, cq1 = {};

#pragma unroll 4
  for (int k = 0; k < Dn; k += 4) {
    v2f a0 = *(const v2f*)(aPtr0 + k);                     // b64
    v2f a1 = *(const v2f*)(aPtr1 + k);                     // b64
    v2f bp, bq;
    bp.x = bPtrP[(size_t)k * Hn];
    bp.y = bPtrP[(size_t)(k + 1) * Hn];
    bq.x = bPtrQ[(size_t)k * Hn];
    bq.y = bPtrQ[(size_t)(k + 1) * Hn];
    // 8 args: (neg_a, A, neg_b, B, c_mod, C, reuse_a, reuse_b)
    cp0 = __builtin_amdgcn_wmma_f32_16x16x4_f32(false, a0, false, bp,
                                                (short)0, cp0, false, false);
    cq0 = __builtin_amdgcn_wmma_f32_16x16x4_f32(false, a0, false, bq,
                                                (short)0, cq0, false, false);
    cp1 = __builtin_amdgcn_wmma_f32_16x16x4_f32(false, a1, false, bp,
                                                (short)0, cp1, false, false);
    cq1 = __builtin_amdgcn_wmma_f32_16x16x4_f32(false, a1, false, bq,
                                                (short)0, cq1, false, false);
  }

  // C/D layout: VGPR r -> row M = tile_base + r + 8*half, col N = n0+(lane&15)
  float* prow0 = P + (size_t)(m0 + 8 * half) * Hn + bcol;
  float* qrow0 = Q + (size_t)(m0 + 8 * half) * Hn + bcol;
  float* prow1 = prow0 + (size_t)16 * Hn;
  float* qrow1 = qrow0 + (size_t)16 * Hn;
#pragma unroll
  for (int r = 0; r < 8; ++r) {
    prow0[(size_t)r * Hn] = cp0[r];
    qrow0[(size_t)r * Hn] = cq0[r];
    prow1[(size_t)r * Hn] = cp1[r];
    qrow1[(size_t)r * Hn] = cq1[r];
  }
}

// ---------------------------------------------------------------------------
// Kernel 2: out[b,t,h] = tanh(P[b, ii[t], h] + Q[b, jj[t], h] + bias[h]).
// Flat index over B*T*(H/4); stores are exactly contiguous v4f and
// non-temporal (228 MB streamed once; P/Q (4.7 MB) stay resident in L2).
// ---------------------------------------------------------------------------
__device__ __forceinline__ int tri_offset(int i) {
  // number of pairs before row i in triu order: i*S - i*(i-1)/2
  return i * Sn - (i * (i - 1)) / 2;
}

__global__ __launch_bounds__(256)
void pair_tanh_epilogue(const float* __restrict__ P,
                        const float* __restrict__ Q,
                        const float* __restrict__ bias,
                        float* __restrict__ out) {
  const int idx = blockIdx.x * 256 + threadIdx.x;   // over Bn*Tn*(Hn/4)
  const int h4 = idx % (Hn / 4);
  const int r  = idx / (Hn / 4);
  const int t  = r % Tn;
  const int b  = r / Tn;
  if (b >= Bn) return;

  // Invert triangular index: largest i with tri_offset(i) <= t.
  const float twoSp1 = (float)(2 * Sn + 1);
  float disc = twoSp1 * twoSp1 - 8.0f * (float)t;
  int i = (int)((twoSp1 - sqrtf(disc)) * 0.5f);
  if (i < 0) i = 0;
  if (i > Sn - 1) i = Sn - 1;
  while (i > 0 && tri_offset(i) > t) --i;                  // exact fix-up
  while (i < Sn - 1 && tri_offset(i + 1) <= t) ++i;
  const int j = i + (t - tri_offset(i));

  const v4f p  = *(const v4f*)(P + ((size_t)b * Sn + i) * Hn + h4 * 4);
  const v4f q  = *(const v4f*)(Q + ((size_t)b * Sn + j) * Hn + h4 * 4);
  const v4f bb = *(const v4f*)(bias + h4 * 4);

  v4f o;
  o.x = tanhf(p.x + q.x + bb.x);
  o.y = tanhf(p.y + q.y + bb.y);
  o.z = tanhf(p.z + q.z + bb.z);
  o.w = tanhf(p.w + q.w + bb.w);

  __builtin_nontemporal_store(o, (v4f*)out + idx);
}

// ---------------------------------------------------------------------------
extern "C" void kernel_launch(void* const* d_in, const int* in_sizes, int n_in,
                              void* d_out, int out_size, void* d_ws, size_t ws_size,
                              hipStream_t stream) {
  const float* X    = (const float*)d_in[0];   // [B, S, D] fp32
  const float* W    = (const float*)d_in[1];   // [2D, H] fp32
  const float* bias = (const float*)d_in[2];   // [H] fp32
  float* out = (float*)d_out;                  // [B, T, H] fp32

  float* P = (float*)d_ws;                     // [Mn, Hn]
  float* Q = P + (size_t)Mn * Hn;              // [Mn, Hn]  (4.7 MB total)

  dim3 gemm_grid(Mn / 32, Hn / 16);            // 48 x 24 tiles, 1 wave each
  pairproj_gemm_wmma<<<gemm_grid, 32, 0, stream>>>(X, W, P, Q);

  const int total4 = Bn * Tn * (Hn / 4);       // 14,229,504 float4 outputs
  const int blocks = (total4 + 255) / 256;
  pair_tanh_epilogue<<<blocks, 256, 0, stream>>>(P, Q, bias, out);
}